// HybridLoss_180388627225
// MI455X (gfx1250) — compile-verified
//
#include <hip/hip_runtime.h>

// Problem constants (match reference)
#define BATCH    524288
#define NSTATE   40
#define NX       20
#define FORCE    8.0f
#define ALPHA_PI 0.1f
#define INV_DT   (1.0f / 0.01f)

// Launch config: BLOCKS * THREADS * ROWS_PER_THREAD == BATCH
#define BLOCKS   1024
#define THREADS  256
#define ROWS_PER_THREAD 2
#define CHUNK    (BLOCKS * THREADS)   // rows per grid sweep

typedef float v2f __attribute__((ext_vector_type(2)));
typedef float v8f __attribute__((ext_vector_type(8)));

// ---------------------------------------------------------------------------
// Exact f32 sum of one value per lane across the wave32, computed with
// V_WMMA_F32_16X16X4_F32.  A-matrix layout (16x4 f32): lane L<16 holds
// A[M=L,K=0] in VGPR0 (K=1 in VGPR1); lane L>=16 holds A[M=L-16,K=2/3].
// With a = {x, 0} per lane and B = all-ones, D[m,n] = x_m + x_{m+16}.
// D layout: lane n<16 holds M=0..7 in its 8 VGPRs, lane n>=16 holds M=8..15,
// so sum(d[0..7]) gives half the total; xor-16 shuffle adds the other half.
// EXEC must be all ones at the call site (no divergence before this).
// ---------------------------------------------------------------------------
__device__ __forceinline__ float wave_sum_wmma(float x) {
    v2f a; a.x = x;    a.y = 0.0f;
    v2f b; b.x = 1.0f; b.y = 1.0f;
    v8f c = {};
    v8f d = __builtin_amdgcn_wmma_f32_16x16x4_f32(
        /*neg_a=*/false, a, /*neg_b=*/false, b,
        /*c_mod=*/(short)0, c, /*reuse_a=*/false, /*reuse_b=*/false);
    float s = ((d[0] + d[1]) + (d[2] + d[3])) + ((d[4] + d[5]) + (d[6] + d[7]));
    s += __shfl_xor(s, 16, 32);   // wave32: combine the two half-sums
    return s;                      // full 32-lane total, in every lane
}

// Block-level reduction of the (sdd, spi) pair. Deterministic order.
__device__ __forceinline__ void block_reduce2(float sdd, float spi,
                                              float* __restrict__ out_dd,
                                              float* __restrict__ out_pi) {
    __shared__ float lds_dd[THREADS / 32];
    __shared__ float lds_pi[THREADS / 32];
    float wdd = wave_sum_wmma(sdd);      // all lanes active here
    float wpi = wave_sum_wmma(spi);
    const int lane = threadIdx.x & 31;
    const int wave = threadIdx.x >> 5;
    if (lane == 0) { lds_dd[wave] = wdd; lds_pi[wave] = wpi; }
    __syncthreads();
    if (threadIdx.x == 0) {
        float a = 0.0f, b = 0.0f;
        #pragma unroll
        for (int w = 0; w < THREADS / 32; ++w) { a += lds_dd[w]; b += lds_pi[w]; }
        *out_dd = a; *out_pi = b;
    }
}

// ---------------------------------------------------------------------------
// Kernel 1: streaming pass. One row (40 f32) per thread per sweep; the
// periodic Lorenz-96 stencil resolves entirely in registers (compile-time
// indices after full unroll). b128 loads; ~210 MB total HBM traffic.
// ---------------------------------------------------------------------------
__global__ __launch_bounds__(THREADS)
void lorenz_loss_partial(const float* __restrict__ pred,
                         const float* __restrict__ targ,
                         const float* __restrict__ ycur,
                         float* __restrict__ partial) {
    const int gid = blockIdx.x * THREADS + threadIdx.x;
    float sdd = 0.0f, spi = 0.0f;

    #pragma unroll
    for (int r = 0; r < ROWS_PER_THREAD; ++r) {
        const int row = gid + r * CHUNK;
        const float4* p4 = (const float4*)(pred + (size_t)row * NSTATE);
        const float4* y4 = (const float4*)(ycur + (size_t)row * NSTATE);
        const float4* t4 = (const float4*)(targ + (size_t)row * NSTATE);

        if (r + 1 < ROWS_PER_THREAD) {
            // gfx1250 global_prefetch_b8 for the next grid sweep
            __builtin_prefetch(pred + (size_t)(row + CHUNK) * NSTATE, 0, 0);
            __builtin_prefetch(ycur + (size_t)(row + CHUNK) * NSTATE, 0, 0);
            __builtin_prefetch(targ + (size_t)(row + CHUNK) * NSTATE, 0, 0);
        }

        float p[NSTATE], y[NSTATE];
        #pragma unroll
        for (int i = 0; i < NSTATE / 4; ++i) {
            float4 v = p4[i];
            p[4*i+0] = v.x; p[4*i+1] = v.y; p[4*i+2] = v.z; p[4*i+3] = v.w;
        }
        #pragma unroll
        for (int i = 0; i < NSTATE / 4; ++i) {
            float4 v = y4[i];
            y[4*i+0] = v.x; y[4*i+1] = v.y; y[4*i+2] = v.z; y[4*i+3] = v.w;
        }

        // Data-driven term: first NX columns only (targets tail never read).
        #pragma unroll
        for (int i = 0; i < NX / 4; ++i) {
            float4 v = t4[i];
            float d0 = p[4*i+0] - v.x;
            float d1 = p[4*i+1] - v.y;
            float d2 = p[4*i+2] - v.z;
            float d3 = p[4*i+3] - v.w;
            sdd += d0*d0 + d1*d1 + d2*d2 + d3*d3;
        }

        // Physics term: f_i = (p[i+1] - p[i-2]) * p[i-1] - p[i] + F (periodic)
        #pragma unroll
        for (int i = 0; i < NSTATE; ++i) {
            float f  = (p[(i + 1) % NSTATE] - p[(i + NSTATE - 2) % NSTATE])
                         * p[(i + NSTATE - 1) % NSTATE]
                       - p[i] + FORCE;
            float dv = (p[i] - y[i]) * INV_DT - f;
            spi += dv * dv;
        }
    }

    block_reduce2(sdd, spi,
                  &partial[2 * blockIdx.x + 0],
                  &partial[2 * blockIdx.x + 1]);
}

// ---------------------------------------------------------------------------
// Kernel 2: single block, fixed-order reduction of 1024 partial pairs,
// then write (total, l_dd, l_pi).
// ---------------------------------------------------------------------------
__global__ __launch_bounds__(THREADS)
void lorenz_loss_final(const float* __restrict__ partial,
                       float* __restrict__ out) {
    float sdd = 0.0f, spi = 0.0f;
    #pragma unroll
    for (int i = 0; i < BLOCKS / THREADS; ++i) {
        const int b = threadIdx.x + i * THREADS;
        sdd += partial[2 * b + 0];
        spi += partial[2 * b + 1];
    }

    __shared__ float tot[2];
    block_reduce2(sdd, spi, &tot[0], &tot[1]);
    __syncthreads();

    if (threadIdx.x == 0) {
        const float l_dd = tot[0] * (1.0f / ((float)BATCH * (float)NX));
        const float l_pi = tot[1] * (1.0f / ((float)BATCH * (float)NSTATE));
        out[0] = l_dd + ALPHA_PI * l_pi;  // total_loss
        out[1] = l_dd;
        out[2] = l_pi;
    }
}

extern "C" void kernel_launch(void* const* d_in, const int* in_sizes, int n_in,
                              void* d_out, int out_size, void* d_ws, size_t ws_size,
                              hipStream_t stream) {
    const float* pred = (const float*)d_in[0];  // predictions
    const float* targ = (const float*)d_in[1];  // targets
    const float* ycur = (const float*)d_in[2];  // y_current_full_state
    float* out      = (float*)d_out;            // (total, l_dd, l_pi)
    float* partials = (float*)d_ws;             // 1024 * 2 floats = 8 KB

    lorenz_loss_partial<<<BLOCKS, THREADS, 0, stream>>>(pred, targ, ycur, partials);
    lorenz_loss_final<<<1, THREADS, 0, stream>>>(partials, out);
}